// LRMCFullnet_9586367004985
// MI455X (gfx1250) — compile-verified
//
#include <hip/hip_runtime.h>
#include <hip/hip_bf16.h>
#include <math.h>

// ---------------------------------------------------------------------------
// LRMC full net on MI455X (gfx1250, wave32, WMMA).
// Heavy ops use V_WMMA_F32_16X16X4_F32 in fused streaming kernels so the
// 64MB Y matrix is read once per pass and no 4096x4096 intermediate is
// ever materialized (HBM-bound: ~1.8GB total @ 23.3TB/s). Y tiles are
// staged into LDS with double-buffered GLOBAL_LOAD_ASYNC_TO_LDS_B128
// (ASYNCcnt-fenced) and prefetched 4 tiles ahead with global_prefetch.
// ---------------------------------------------------------------------------

typedef __attribute__((ext_vector_type(2))) float v2f;
typedef __attribute__((ext_vector_type(8))) float v8f;

#define NDIM 4096
#define RNK  20
#define RP   32   // padded row stride for rank-20 factors (cols 20..31 == 0)

// D(16x16,f32) = A(16x4,f32) * B(4x16,f32) + C   -- wave32, VOP3P
__device__ __forceinline__ v8f wmma_f32x4(v2f a, v2f b, v8f c) {
    return __builtin_amdgcn_wmma_f32_16x16x4_f32(
        /*neg_a=*/false, a, /*neg_b=*/false, b,
        /*c_mod=*/(short)0, c, /*reuse_a=*/false, /*reuse_b=*/false);
}

// LDS byte offset of a generic pointer that points into shared memory
__device__ __forceinline__ unsigned lds_off(const void* p) {
    return (unsigned)(size_t)(__attribute__((address_space(3))) const void*)p;
}

__device__ __forceinline__ void wait_async_le2() {
    asm volatile("s_wait_asynccnt 0x2" ::: "memory");
}
__device__ __forceinline__ void wait_async_0() {
    asm volatile("s_wait_asynccnt 0x0" ::: "memory");
}

// Issue async copy of one 16x16 f32 tile of Y (memory order) into LDS.
// Two b128 ops: 32 lanes x 4 floats = 8 rows x 16 cols each.
template <int TRANSP>
__device__ __forceinline__ void issue_tile_async(const float* __restrict__ Y,
                                                 unsigned ldsbase,
                                                 int a0, int b0, int lane) {
    const int row  = lane >> 2;        // 0..7
    const int col4 = (lane & 3) * 4;   // 0,4,8,12
#pragma unroll
    for (int inst = 0; inst < 2; ++inst) {
        const int r = row + 8 * inst;
        const float* gp = TRANSP ? (Y + (size_t)(b0 + r) * NDIM + a0 + col4)
                                 : (Y + (size_t)(a0 + r) * NDIM + b0 + col4);
        const unsigned loff = ldsbase + (unsigned)((r * 16 + col4) * 4);
        asm volatile("global_load_async_to_lds_b128 %0, %1, off"
                     :: "v"(loff), "v"((unsigned long long)(size_t)gp)
                     : "memory");
    }
}

// ---------------------------------------------------------------------------
// Generic fused tall-skinny GEMM over a virtual matrix Fv (4096x4096):
//   MODE 0: Fv[a,b] =  clamp(Yv[a,b], +-z)                (== Y - soft(Y,z))
//   MODE 1: Fv[a,b] = -clamp(Yv[a,b] - Lm[a,:].Rm[b,:], +-z)
//                                                         (== soft(d,z)-d)
//   Yv = TRANSP ? Y^T : Y
// Computes  Gout[4096x32] += Fv @ Bin[4096x32]  (rank cols 0..19, pad 0).
// Grid: 512 blocks x 128 thr = 2048 waves; wave = (strip of 16 rows, 32
// column-chunks of 16). 13 WMMAs per 16x16 tile in MODE 1.
// ---------------------------------------------------------------------------
template <int MODE, int TRANSP>
__global__ __launch_bounds__(128) void tsgemm_kernel(
    const float* __restrict__ Y, const float* __restrict__ Lm,
    const float* __restrict__ Rm, const float* __restrict__ Bin,
    float* __restrict__ Gout, const float* __restrict__ zeta, int zidx)
{
    __shared__ float ldsT[4][16 * 16];       // L.R^T tile (MODE 1 only)
    __shared__ float ldsY[4][2][16 * 16];    // double-buffered Y tile
    const int lane = threadIdx.x & 31;
    const int w    = threadIdx.x >> 5;
    const int wid  = blockIdx.x * 4 + w;     // 0..2047
    const int strip = wid >> 3;              // 0..255
    const int wseg  = wid & 7;               // 0..7
    const int a0 = strip * 16;
    const int m16 = lane & 15;
    const int hi  = lane >> 4;               // 0 | 1
    const float z = zeta[zidx];
    float* myT = ldsT[w];
    const unsigned ybase0 = lds_off(&ldsY[w][0][0]);

    // Hoist the loop-invariant 16x20 strip of Lm (A operands for stage 1)
    v2f aL[5];
    if (MODE == 1) {
        #pragma unroll
        for (int kk = 0; kk < 5; ++kk)
            aL[kk] = *(const v2f*)(Lm + (size_t)(a0 + m16) * RP + kk * 4 + 2 * hi);
    }

    v8f acc0 = {};  // cols 0..15 of Gout strip
    v8f acc1 = {};  // cols 16..31

    const int cb0 = wseg * 32;
    issue_tile_async<TRANSP>(Y, ybase0, a0, cb0 * 16, lane);  // prologue

    for (int i = 0; i < 32; ++i) {
        const int cb = cb0 + i;
        const int b0 = cb * 16;
        const float* yt = &ldsY[w][i & 1][0];

        if (i < 31)   // stage next tile while we compute this one
            issue_tile_async<TRANSP>(Y, ybase0 + ((unsigned)(~i & 1)) * 1024,
                                     a0, (cb + 1) * 16, lane);
        if (i + 4 < 32) {   // deep prefetch: one cacheline per row, 4 tiles out
            const int bp = (cb + 4) * 16;
            const float* pf = TRANSP ? (Y + (size_t)(bp + m16) * NDIM + a0)
                                     : (Y + (size_t)(a0 + m16) * NDIM + bp);
            __builtin_prefetch(pf, 0, 1);
        }

        if (MODE == 1) {
            // T = Lm_strip(16x20) @ Rm_chunk^T(20x16) via 5 K=4 WMMAs
            v8f t = {};
            #pragma unroll
            for (int kk = 0; kk < 5; ++kk) {
                v2f b = *(const v2f*)(Rm + (size_t)(b0 + m16) * RP + kk * 4 + 2 * hi);
                t = wmma_f32x4(aL[kk], b, t);
            }
            // stash T (C-layout) row-major into LDS for A-layout reload
            #pragma unroll
            for (int v = 0; v < 8; ++v)
                myT[(v + 8 * hi) * 16 + m16] = t[v];
            __syncthreads();
        }

        if (i < 31) wait_async_le2(); else wait_async_0();

        // Stage 2: Gacc += Fv_tile(16x16) @ Bin_chunk(16x32)
        #pragma unroll
        for (int kc = 0; kc < 4; ++kc) {
            const int c0 = kc * 4 + 2 * hi;  // column-pair inside tile
            float y0, y1;
            if (TRANSP) {                    // tile stored in memory order (b,a)
                y0 = yt[c0 * 16 + m16];
                y1 = yt[(c0 + 1) * 16 + m16];
            } else {                         // tile stored (a,b) row-major
                v2f yy = *(const v2f*)(yt + m16 * 16 + c0);
                y0 = yy.x; y1 = yy.y;
            }
            v2f aop;
            if (MODE == 1) {
                v2f tv = *(const v2f*)(myT + m16 * 16 + c0);
                float d0 = y0 - tv.x, d1 = y1 - tv.y;
                aop.x = -fminf(fmaxf(d0, -z), z);
                aop.y = -fminf(fmaxf(d1, -z), z);
            } else {
                aop.x = fminf(fmaxf(y0, -z), z);
                aop.y = fminf(fmaxf(y1, -z), z);
            }
            // B operand rows b0+kc*4+2*hi, +1 of Bin
            const float* bp = Bin + (size_t)(b0 + kc * 4 + 2 * hi) * RP;
            v2f bv0; bv0.x = bp[m16];      bv0.y = bp[RP + m16];
            v2f bv1; bv1.x = bp[16 + m16]; bv1.y = bp[RP + 16 + m16];
            acc0 = wmma_f32x4(aop, bv0, acc0);
            acc1 = wmma_f32x4(aop, bv1, acc1);
        }
        if (MODE == 1) __syncthreads();
    }

    // 8 waves per strip -> accumulate with float atomics
    #pragma unroll
    for (int v = 0; v < 8; ++v) {
        const int row = a0 + v + 8 * hi;
        atomicAdd(&Gout[(size_t)row * RP + m16],      acc0[v]);
        atomicAdd(&Gout[(size_t)row * RP + 16 + m16], acc1[v]);
    }
}

// ---------------------------------------------------------------------------
// Fused loss: sum((U V^T - L R^T)^2) and sum((U V^T)^2) over 4096^2,
// both products via WMMA f32. 2048 blocks x 128 -> 8192 waves x 8 tiles.
// A wave's 8 tiles share i0, so U/L A-operands are hoisted once per wave.
// ---------------------------------------------------------------------------
__global__ __launch_bounds__(128) void loss_kernel(
    const float* __restrict__ U, const float* __restrict__ V,
    const float* __restrict__ L, const float* __restrict__ R,
    float* __restrict__ scal)
{
    const int lane = threadIdx.x & 31;
    const int w    = threadIdx.x >> 5;
    const int wid  = blockIdx.x * 4 + w;        // 0..8191
    const int m16  = lane & 15, hi = lane >> 4;
    const int i0   = ((wid * 8) >> 8) * 16;     // constant across the 8 tiles
    float sn = 0.f, sd = 0.f;

    v2f aU[5], aL[5];
    #pragma unroll
    for (int kk = 0; kk < 5; ++kk) {
        aU[kk] = *(const v2f*)(U + (size_t)(i0 + m16) * RNK + kk * 4 + 2 * hi);
        aL[kk] = *(const v2f*)(L + (size_t)(i0 + m16) * RP  + kk * 4 + 2 * hi);
    }

    for (int k = 0; k < 8; ++k) {
        const int tix = wid * 8 + k;            // 0..65535
        const int j0 = (tix & 255) * 16;
        if (k + 2 < 8) {                        // prefetch B-operand rows ahead
            const int jp = ((tix + 2) & 255) * 16;
            __builtin_prefetch(V + (size_t)(jp + m16) * RNK, 0, 1);
            __builtin_prefetch(R + (size_t)(jp + m16) * RP, 0, 1);
        }
        v8f x = {}, p = {};
        #pragma unroll
        for (int kk = 0; kk < 5; ++kk) {
            v2f bv = *(const v2f*)(V + (size_t)(j0 + m16) * RNK + kk * 4 + 2 * hi);
            x = wmma_f32x4(aU[kk], bv, x);
            v2f br = *(const v2f*)(R + (size_t)(j0 + m16) * RP + kk * 4 + 2 * hi);
            p = wmma_f32x4(aL[kk], br, p);
        }
        #pragma unroll
        for (int v = 0; v < 8; ++v) {
            float d = x[v] - p[v];
            sn += d * d;
            sd += x[v] * x[v];
        }
    }
    atomicAdd(&scal[0], sn);
    atomicAdd(&scal[1], sd);
}

// ---------------------------------------------------------------------------
// Small helpers (20x20 algebra, fills, RNG sketch, final scalar)
// ---------------------------------------------------------------------------
__global__ void fill_kernel(float* p, int n, float v) {
    int i = blockIdx.x * blockDim.x + threadIdx.x;
    if (i < n) p[i] = v;
}

__device__ __forceinline__ unsigned hashu(unsigned x) {
    x ^= x >> 17; x *= 0xed5ad4bbu; x ^= x >> 11; x *= 0xac4c1b51u;
    x ^= x >> 15; x *= 0x31848babu; x ^= x >> 14; return x;
}

__global__ void ginit_kernel(float* G) {   // Gaussian sketch, padded [4096x32]
    int idx = blockIdx.x * blockDim.x + threadIdx.x;   // 4096*32
    int n = idx & 31;
    float v = 0.f;
    if (n < RNK) {
        unsigned h1 = hashu(2u * idx + 1u), h2 = hashu(2u * idx + 2u);
        float u1 = (h1 + 1.0f) * 2.3283064e-10f;       // (0,1]
        float u2 = h2 * 2.3283064e-10f;
        v = sqrtf(-2.f * logf(u1)) * __cosf(6.2831853f * u2);
    }
    G[idx] = v;
}

// W(20x20) += X_slab^T X_slab ; grid 16 x 256, W pre-zeroed
__global__ void gram_kernel(const float* __restrict__ X, float* __restrict__ W) {
    const int t = threadIdx.x;
    const int i0 = blockIdx.x * 256;
    for (int e = t; e < RNK * RNK; e += 256) {
        const int a = e / RNK, b = e % RNK;
        float s = 0.f;
        for (int i = i0; i < i0 + 256; ++i)
            s += X[(size_t)i * RP + a] * X[(size_t)i * RP + b];
        atomicAdd(&W[e], s);
    }
}

// Cholesky-QR prep: M = inv(chol(W))^T so that Q = X @ M is orthonormal
__global__ void cholqr_prep_kernel(const float* __restrict__ W, float* __restrict__ M) {
    if (threadIdx.x != 0) return;
    float C[RNK][RNK], Ci[RNK][RNK];
    for (int i = 0; i < RNK; ++i)
        for (int j = 0; j < RNK; ++j) { C[i][j] = 0.f; Ci[i][j] = 0.f; }
    for (int i = 0; i < RNK; ++i)
        for (int j = 0; j <= i; ++j) {
            float s = W[i * RNK + j];
            for (int k = 0; k < j; ++k) s -= C[i][k] * C[j][k];
            if (i == j) C[i][j] = sqrtf(fmaxf(s, 1e-20f));
            else        C[i][j] = s / C[j][j];
        }
    for (int i = 0; i < RNK; ++i) {
        Ci[i][i] = 1.f / C[i][i];
        for (int j = 0; j < i; ++j) {
            float s = 0.f;
            for (int k = j; k < i; ++k) s += C[i][k] * Ci[k][j];
            Ci[i][j] = -s / C[i][i];
        }
    }
    for (int k = 0; k < RNK; ++k)
        for (int n = 0; n < RNK; ++n) M[k * RNK + n] = Ci[n][k];
}

// Jacobi eig of SPD C20 = B B^T = Ub diag(sig^2) Ub^T ; emits
//   M1 = Ub * diag(sig^0.5)   (L = Q @ M1)
//   M2 = Ub * diag(sig^-0.5)  (R = B^T @ M2)
__global__ void eig_prep_kernel(const float* __restrict__ C20,
                                float* __restrict__ M1, float* __restrict__ M2) {
    if (threadIdx.x != 0) return;
    float A[RNK][RNK], Vv[RNK][RNK];
    for (int i = 0; i < RNK; ++i)
        for (int j = 0; j < RNK; ++j) {
            A[i][j] = C20[i * RNK + j];
            Vv[i][j] = (i == j) ? 1.f : 0.f;
        }
    for (int sweep = 0; sweep < 30; ++sweep)
        for (int p = 0; p < RNK - 1; ++p)
            for (int q = p + 1; q < RNK; ++q) {
                float apq = A[p][q];
                if (fabsf(apq) < 1e-18f) continue;
                float theta = 0.5f * (A[q][q] - A[p][p]) / apq;
                float t = (theta >= 0.f ? 1.f : -1.f) /
                          (fabsf(theta) + sqrtf(1.f + theta * theta));
                float c = 1.f / sqrtf(1.f + t * t), s = t * c;
                for (int k = 0; k < RNK; ++k) {
                    float akp = A[k][p], akq = A[k][q];
                    A[k][p] = c * akp - s * akq;
                    A[k][q] = s * akp + c * akq;
                }
                for (int k = 0; k < RNK; ++k) {
                    float apk = A[p][k], aqk = A[q][k];
                    A[p][k] = c * apk - s * aqk;
                    A[q][k] = s * apk + c * aqk;
                }
                for (int k = 0; k < RNK; ++k) {
                    float vkp = Vv[k][p], vkq = Vv[k][q];
                    Vv[k][p] = c * vkp - s * vkq;
                    Vv[k][q] = s * vkp + c * vkq;
                }
            }
    for (int n = 0; n < RNK; ++n) {
        float d = fmaxf(A[n][n], 1e-12f);
        float s4 = sqrtf(sqrtf(d));            // sqrt(Sigma0)
        for (int k = 0; k < RNK; ++k) {
            M1[k * RNK + n] = Vv[k][n] * s4;
            M2[k * RNK + n] = Vv[k][n] / s4;
        }
    }
}

// Gauss-Jordan inverse of SPD 20x20
__global__ void inv20_kernel(const float* __restrict__ Ain, float* __restrict__ Aout) {
    if (threadIdx.x != 0) return;
    float M[RNK][2 * RNK];
    for (int i = 0; i < RNK; ++i)
        for (int j = 0; j < RNK; ++j) {
            M[i][j] = Ain[i * RNK + j];
            M[i][RNK + j] = (i == j) ? 1.f : 0.f;
        }
    for (int c = 0; c < RNK; ++c) {
        float piv = M[c][c];
        if (fabsf(piv) < 1e-30f) piv = 1e-30f;
        float ip = 1.f / piv;
        for (int j = 0; j < 2 * RNK; ++j) M[c][j] *= ip;
        for (int r = 0; r < RNK; ++r) {
            if (r == c) continue;
            float f = M[r][c];
            for (int j = 0; j < 2 * RNK; ++j) M[r][j] -= f * M[c][j];
        }
    }
    for (int i = 0; i < RNK; ++i)
        for (int j = 0; j < RNK; ++j) Aout[i * RNK + j] = M[i][RNK + j];
}

// Out[4096x32] = In[4096x32(cols<20)] @ M(20x20); zero-padded cols
__global__ void apply_right_kernel(const float* __restrict__ In,
                                   const float* __restrict__ M,
                                   float* __restrict__ Out) {
    const int i = blockIdx.x * blockDim.x + threadIdx.x;   // 4096
    float r[RNK];
    #pragma unroll
    for (int k = 0; k < RNK; ++k) r[k] = In[(size_t)i * RP + k];
    #pragma unroll
    for (int n = 0; n < RNK; ++n) {
        float s = 0.f;
        #pragma unroll
        for (int k = 0; k < RNK; ++k) s += r[k] * M[k * RNK + n];
        Out[(size_t)i * RP + n] = s;
    }
    #pragma unroll
    for (int n = RNK; n < RP; ++n) Out[(size_t)i * RP + n] = 0.f;
}

// Lnew = Lc - eta[t] * (G @ Minv)
__global__ void update_kernel(const float* __restrict__ Lc,
                              const float* __restrict__ G,
                              const float* __restrict__ Minv,
                              const float* __restrict__ eta, int t,
                              float* __restrict__ Ln) {
    const int i = blockIdx.x * blockDim.x + threadIdx.x;   // 4096
    const float e = eta[t];
    float g[RNK];
    #pragma unroll
    for (int k = 0; k < RNK; ++k) g[k] = G[(size_t)i * RP + k];
    #pragma unroll
    for (int n = 0; n < RNK; ++n) {
        float s = 0.f;
        #pragma unroll
        for (int k = 0; k < RNK; ++k) s += g[k] * Minv[k * RNK + n];
        Ln[(size_t)i * RP + n] = Lc[(size_t)i * RP + n] - e * s;
    }
    #pragma unroll
    for (int n = RNK; n < RP; ++n) Ln[(size_t)i * RP + n] = 0.f;
}

__global__ void final_kernel(const float* __restrict__ scal, float* __restrict__ out) {
    if (threadIdx.x == 0 && blockIdx.x == 0)
        out[0] = sqrtf(scal[0] / scal[1]);
}

// ---------------------------------------------------------------------------
// Orchestration
// ---------------------------------------------------------------------------
extern "C" void kernel_launch(void* const* d_in, const int* in_sizes, int n_in,
                              void* d_out, int out_size, void* d_ws, size_t ws_size,
                              hipStream_t stream) {
    (void)in_sizes; (void)n_in; (void)out_size; (void)ws_size;
    const float* U    = (const float*)d_in[0];   // [4096,20]
    const float* V    = (const float*)d_in[1];   // [4096,20]
    const float* Y    = (const float*)d_in[2];   // [4096,4096]
    const float* zeta = (const float*)d_in[4];   // [10]
    const float* eta  = (const float*)d_in[5];   // [10]
    float* out = (float*)d_out;

    const size_t NM = (size_t)NDIM * RP;         // 131072 floats per factor buf
    float* ws = (float*)d_ws;
    float* G    = ws;            ws += NM;
    float* Tmp  = ws;            ws += NM;
    float* Qb   = ws;            ws += NM;
    float* Bt   = ws;            ws += NM;
    float* L0   = ws;            ws += NM;
    float* R0   = ws;            ws += NM;
    float* L1   = ws;            ws += NM;
    float* R1   = ws;            ws += NM;
    float* GL   = ws;            ws += NM;
    float* GR   = ws;            ws += NM;
    float* W    = ws;            ws += 512;
    float* M1   = ws;            ws += 512;
    float* M2   = ws;            ws += 512;
    float* RtR  = ws;            ws += 512;
    float* LtL  = ws;            ws += 512;
    float* RRi  = ws;            ws += 512;
    float* LLi  = ws;            ws += 512;
    float* scal = ws;            ws += 8;

    auto zero = [&](float* p, size_t n) {
        fill_kernel<<<(unsigned)((n + 255) / 256), 256, 0, stream>>>(p, (int)n, 0.f);
    };
    auto ts = [&](const float* Lm, const float* Rm, const float* Bin, float* Gout,
                  int zidx, int mode, int transp) {
        if (mode == 0 && transp == 0)
            tsgemm_kernel<0, 0><<<512, 128, 0, stream>>>(Y, Lm, Rm, Bin, Gout, zeta, zidx);
        else if (mode == 0)
            tsgemm_kernel<0, 1><<<512, 128, 0, stream>>>(Y, Lm, Rm, Bin, Gout, zeta, zidx);
        else if (transp == 0)
            tsgemm_kernel<1, 0><<<512, 128, 0, stream>>>(Y, Lm, Rm, Bin, Gout, zeta, zidx);
        else
            tsgemm_kernel<1, 1><<<512, 128, 0, stream>>>(Y, Lm, Rm, Bin, Gout, zeta, zidx);
    };
    auto cholqr = [&](const float* X, float* Qout) {   // Qout = orth(X)
        zero(W, 400);
        gram_kernel<<<16, 256, 0, stream>>>(X, W);
        cholqr_prep_kernel<<<1, 32, 0, stream>>>(W, M1);
        apply_right_kernel<<<16, 256, 0, stream>>>(X, M1, Qout);
    };

    // ---- iteration 0: A = clamp(Y, +-zeta[0]); randomized rank-20 SVD ----
    ginit_kernel<<<512, 256, 0, stream>>>(G);
    zero(Tmp, NM);
    ts(Y, Y, G, Tmp, 0, /*mode*/0, /*transp*/0);       // Tmp = A @ G
    cholqr(Tmp, Qb);                                   // Q = qr(A G)
    for (int it = 0; it < 4; ++it) {                   // power iterations
        zero(Tmp, NM); ts(Y, Y, Qb, Tmp, 0, 0, 1);     // Tmp = A^T Q
        cholqr(Tmp, Qb);
        zero(Tmp, NM); ts(Y, Y, Qb, Tmp, 0, 0, 0);     // Tmp = A Q
        cholqr(Tmp, Qb);
    }
    zero(Bt, NM);
    ts(Y, Y, Qb, Bt, 0, 0, 1);                         // B^T = A^T Q
    zero(W, 400);
    gram_kernel<<<16, 256, 0, stream>>>(Bt, W);        // C = B B^T
    eig_prep_kernel<<<1, 32, 0, stream>>>(W, M1, M2);
    apply_right_kernel<<<16, 256, 0, stream>>>(Qb, M1, L0);  // L = Q Ub sig^.5
    apply_right_kernel<<<16, 256, 0, stream>>>(Bt, M2, R0);  // R = B^T Ub sig^-.5

    // ---- proximal-gradient iterations t = 1..9 (fused streaming) ----
    float *Lc = L0, *Rc = R0, *Ln = L1, *Rn = R1;
    for (int t = 1; t < 10; ++t) {
        zero(GL, NM); zero(GR, NM);
        ts(Lc, Rc, Rc, GL, t, /*mode*/1, /*transp*/0); // GL = mid @ R
        ts(Rc, Lc, Lc, GR, t, /*mode*/1, /*transp*/1); // GR = mid^T @ L
        zero(RtR, 400); gram_kernel<<<16, 256, 0, stream>>>(Rc, RtR);
        zero(LtL, 400); gram_kernel<<<16, 256, 0, stream>>>(Lc, LtL);
        inv20_kernel<<<1, 32, 0, stream>>>(RtR, RRi);
        inv20_kernel<<<1, 32, 0, stream>>>(LtL, LLi);
        update_kernel<<<16, 256, 0, stream>>>(Lc, GL, RRi, eta, t, Ln);
        update_kernel<<<16, 256, 0, stream>>>(Rc, GR, LLi, eta, t, Rn);
        float* sw;
        sw = Lc; Lc = Ln; Ln = sw;
        sw = Rc; Rc = Rn; Rn = sw;
    }

    // ---- loss = ||U V^T - L R^T||_F / ||U V^T||_F (fully fused) ----
    zero(scal, 2);
    loss_kernel<<<2048, 128, 0, stream>>>(U, V, Lc, Rc, scal);
    final_kernel<<<1, 32, 0, stream>>>(scal, out);
}